// End2EndModel_10694468567068
// MI455X (gfx1250) — compile-verified
//
#include <hip/hip_runtime.h>

typedef __attribute__((ext_vector_type(2))) float v2f;
typedef __attribute__((ext_vector_type(8))) float v8f;

#define BATCH   64
#define NQ      300
#define NCH     84      // 4 + 80
#define NCLS    80
#define TOPK    150
#define KDIM    1200    // 3*20*20
#define NDIM    25200   // 300*84
#define IMGW    640

// ---------------------------------------------------------------------------
// Kernel 1: 32x32 average pool of int32 image (with /255 and BGR swap).
// One wave32 per pooled output element: lane j sums column j over 32 rows
// (fully coalesced 128B lines), integer accumulate, wave-reduce, scale.
// ---------------------------------------------------------------------------
__global__ void pool_kernel(const int* __restrict__ x, float* __restrict__ pooled) {
    const int wave = threadIdx.x >> 5;
    const int lane = threadIdx.x & 31;
    const int o = blockIdx.x * 8 + wave;          // 0 .. 76799 (64*1200)
    const int b    = o / KDIM;
    const int r    = o % KDIM;
    const int cNew = r / 400;                     // pooled channel (BGR order)
    const int cOld = 2 - cNew;                    // source channel
    const int s    = r % 400;
    const int ph   = s / 20;
    const int pw   = s % 20;

    const int* base = x + (((size_t)b * 3 + cOld) * IMGW + (size_t)ph * 32) * IMGW
                        + pw * 32 + lane;
    int sum = 0;
    #pragma unroll
    for (int i = 0; i < 32; ++i)
        sum += base[(size_t)i * IMGW];

    // wave32 reduction
    #pragma unroll
    for (int off = 16; off > 0; off >>= 1)
        sum += __shfl_xor(sum, off, 32);

    if (lane == 0)
        pooled[o] = (float)sum * (1.0f / (255.0f * 1024.0f));
}

// ---------------------------------------------------------------------------
// Kernel 2: (64 x 1200) @ (1200 x 25200) with V_WMMA_F32_16X16X4_F32.
// Block = 128 threads = 4 waves; wave w owns M-tile [16w,16w+16), block owns
// one 16-wide N-tile. K loop steps by 4 (WMMA K), 300 wmma per wave.
// A frag (16x4 f32, 2 VGPR): lanes 0-15 hold K={k0,k0+1}, lanes 16-31 hold
// K={k0+2,k0+3}, M = lane&15. B frag mirrors with N = lane&15.
// C/D (8 VGPR): row = v + (lane>=16 ? 8 : 0), col = lane&15.
// ---------------------------------------------------------------------------
__global__ void gemm_wmma(const float* __restrict__ A,
                          const float* __restrict__ Wm,
                          float* __restrict__ Y) {
    const int lane  = threadIdx.x & 31;
    const int m0    = (threadIdx.x >> 5) << 4;    // 0,16,32,48
    const int n0    = blockIdx.x << 4;            // N tile base (1575 tiles)
    const int mrow  = m0 + (lane & 15);
    const int khalf = (lane >> 4) << 1;           // 0 or 2
    const int col   = n0 + (lane & 15);

    v8f acc = {};
    for (int k0 = 0; k0 < KDIM; k0 += 4) {
        const int kb = k0 + khalf;
        v2f a = *(const v2f*)(A + (size_t)mrow * KDIM + kb);   // 8B aligned
        v2f bfrag;
        bfrag.x = Wm[(size_t)kb * NDIM + col];
        bfrag.y = Wm[(size_t)(kb + 1) * NDIM + col];
        acc = __builtin_amdgcn_wmma_f32_16x16x4_f32(
            /*neg_a=*/false, a, /*neg_b=*/false, bfrag,
            /*c_mod=*/(short)0, acc, /*reuse_a=*/false, /*reuse_b=*/false);
    }

    const int rbase = m0 + ((lane >> 4) << 3);
    #pragma unroll
    for (int v = 0; v < 8; ++v)
        Y[(size_t)(rbase + v) * NDIM + col] = acc[v];
}

// ---------------------------------------------------------------------------
// Kernel 3: per-query class max/argmax, then stable top-150 per batch via
// rank counting (matches jax.lax.top_k descending order + index tie-break),
// then gather [box(4), score, class_id].
// One block per batch, 320 threads (thread q handles query q).
// ---------------------------------------------------------------------------
__global__ void topk_kernel(const float* __restrict__ Y, float* __restrict__ out) {
    const int b = blockIdx.x;
    const int q = threadIdx.x;

    __shared__ float sc[NQ];

    float m = 0.0f;
    int   id = 0;
    const float* row = Y + ((size_t)b * NQ + q) * NCH;
    if (q < NQ) {
        m = row[4];
        #pragma unroll 4
        for (int c = 1; c < NCLS; ++c) {
            float v = row[4 + c];
            if (v > m) { m = v; id = c; }   // strict > : first-occurrence argmax
        }
        sc[q] = m;
    }
    __syncthreads();

    if (q < NQ) {
        int rank = 0;
        for (int j = 0; j < NQ; ++j) {
            float sj = sc[j];
            rank += (sj > m) || (sj == m && j < q);
        }
        if (rank < TOPK) {
            float* o = out + ((size_t)b * TOPK + rank) * 6;
            o[0] = row[0];
            o[1] = row[1];
            o[2] = row[2];
            o[3] = row[3];
            o[4] = m;
            o[5] = (float)id;
        }
    }
}

// ---------------------------------------------------------------------------
extern "C" void kernel_launch(void* const* d_in, const int* in_sizes, int n_in,
                              void* d_out, int out_size, void* d_ws, size_t ws_size,
                              hipStream_t stream) {
    const int*   x  = (const int*)d_in[0];     // (64,3,640,640) int32
    const float* Wm = (const float*)d_in[1];   // (1200, 25200) f32
    float* out = (float*)d_out;                // (64,150,6) f32

    char* ws = (char*)d_ws;
    float* pooled = (float*)ws;                                   // 64*1200 f32
    size_t poolBytes = (size_t)BATCH * KDIM * sizeof(float);
    float* Y = (float*)(ws + ((poolBytes + 255) & ~(size_t)255)); // 64*25200 f32

    // 76800 pooled outputs, 8 waves (outputs) per 256-thread block
    pool_kernel<<<(BATCH * KDIM) / 8, 256, 0, stream>>>(x, pooled);

    // 25200/16 = 1575 N-tiles; 4 waves cover M=64
    gemm_wmma<<<NDIM / 16, 128, 0, stream>>>(pooled, Wm, Y);

    topk_kernel<<<BATCH, 320, 0, stream>>>(Y, out);
}